// DeepInterestNet_80633716015461
// MI455X (gfx1250) — compile-verified
//
#include <hip/hip_runtime.h>

// DeepInterestNet — optimized for MI455X (gfx1250, wave32).
//
// Algebraic simplification: softmax over the size-1 last axis is identically 1,
// so the 4-layer MLP / mask / NEG are dead code and the output reduces to
//     out[b,e] = sum_l seq_fea_val[b,l] * emb_table[seq_fea[b,l], e]
// This is a pure gather-scale-sum: ~52 MB of gather traffic over a 25.6 MB
// embedding table that fits in the 192 MB L2 -> HBM-bound at ~27 MB (~1.2 us).
//
// Cross-wave reduction of the 8 per-wave fp32 partials is done on the matrix
// unit with V_WMMA_F32_16X16X4_F32 using an all-ones A matrix (exact fp32
// column sums, layout-permutation invariant).

typedef __attribute__((ext_vector_type(2))) float v2f;
typedef __attribute__((ext_vector_type(8))) float v8f;

#define B_    1024
#define L_    200
#define E_    64
#define WAVES 8
#define LPW   25   // L_ / WAVES

__global__ __launch_bounds__(256) void din_gather_sum_kernel(
    const int*   __restrict__ seq_fea,   // [B,L]
    const float* __restrict__ seq_val,   // [B,L]
    const float* __restrict__ emb,       // [V,E]
    float*       __restrict__ out)       // [B,E]
{
    __shared__ float lds[WAVES * E_];    // per-wave partial sums, 2 KB

    const int tid  = threadIdx.x;
    const int lane = tid & 31;
    const int w    = tid >> 5;
    const int b    = blockIdx.x;

    // Each wave owns 25 sequence positions; lanes 0..24 stage (idx, val).
    const int pos = w * LPW + lane;
    int   idx = 0;
    float val = 0.0f;
    if (lane < LPW) {
        idx = seq_fea[b * L_ + pos];
        val = seq_val[b * L_ + pos];
        // gfx1250 global_prefetch_b8: warm the embedding row into L2/L0.
        __builtin_prefetch(emb + (size_t)idx * E_, 0, 0);
    }

    // Accumulate: lane owns embedding columns (2*lane, 2*lane+1).
    float accx = 0.0f, accy = 0.0f;
    #pragma unroll
    for (int j = 0; j < LPW; ++j) {
        const int   ij = __shfl(idx, j, 32);   // v_readlane broadcast
        const float vj = __shfl(val, j, 32);
        const float2* row = (const float2*)(emb + (size_t)ij * E_);
        const float2 e = row[lane];            // coalesced 256B/wave (b64)
        accx = fmaf(vj, e.x, accx);
        accy = fmaf(vj, e.y, accy);
    }

    // Wave w's partial vector P_w[64] -> LDS (float2 keeps column order).
    ((float2*)lds)[w * 32 + lane] = make_float2(accx, accy);
    __syncthreads();

    // Wave 0: reduce the 8 partial vectors with the matrix unit.
    // A = 16x4 all-ones (fp32), so D[:,n] = sum_k B[k,n] + C[:,n] exactly,
    // independent of the K-row permutation of the B layout. Within a B VGPR,
    // lane l supplies column (l & 15); the two halves and the two VGPR slots
    // are four distinct K rows, so one WMMA sums 4 wave-partials per column,
    // and chaining C accumulates all 8. Four 16-column tiles cover E=64.
    if (tid < 32) {                      // full wave 0: EXEC = all ones
        v2f a; a.x = 1.0f; a.y = 1.0f;   // A: 16x4 ones
        const int j16 = lane & 15;
        const int h   = lane >> 4;       // which half of the wave
        #pragma unroll
        for (int t = 0; t < 4; ++t) {
            const int c = t * 16 + j16;  // output column for this lane
            v2f b1, b2;
            b1.x = lds[(0 + h) * E_ + c];   // waves 0,1
            b1.y = lds[(2 + h) * E_ + c];   // waves 2,3
            b2.x = lds[(4 + h) * E_ + c];   // waves 4,5
            b2.y = lds[(6 + h) * E_ + c];   // waves 6,7
            v8f d = {};
            d = __builtin_amdgcn_wmma_f32_16x16x4_f32(
                    false, a, false, b1, (short)0, d, false, false);
            d = __builtin_amdgcn_wmma_f32_16x16x4_f32(
                    false, a, false, b2, (short)0, d, false, false);
            // Every lane's d[0] holds column (lane&15) of this tile
            // (all D rows are identical since A's rows are identical).
            if (lane < 16) out[b * E_ + c] = d[0];
        }
    }
}

extern "C" void kernel_launch(void* const* d_in, const int* in_sizes, int n_in,
                              void* d_out, int out_size, void* d_ws, size_t ws_size,
                              hipStream_t stream)
{
    // setup_inputs() order:
    // 0 seq_fea [B,L] i32, 1 seq_fea_val [B,L] f32, 2 seq_dft_fea (unused),
    // 3 target_ad_fea_emb (dead), 4 emb_table [V,E] f32, 5..12 MLP weights (dead).
    const int*   seq_fea = (const int*)  d_in[0];
    const float* seq_val = (const float*)d_in[1];
    const float* emb     = (const float*)d_in[4];
    float*       out     = (float*)      d_out;   // [B,E] = 1024x64 f32

    din_gather_sum_kernel<<<dim3(B_), dim3(256), 0, stream>>>(
        seq_fea, seq_val, emb, out);
}